// TriggerGenerator_66889820668158
// MI455X (gfx1250) — compile-verified
//
#include <hip/hip_runtime.h>
#include <math.h>

#define FEAT 256
#define HID 64
#define ONODES 2048
#define TEMPL 10

typedef __attribute__((ext_vector_type(2))) float v2f;
typedef __attribute__((ext_vector_type(4))) float v4f;
typedef __attribute__((ext_vector_type(8))) float v8f;

// ---------------------------------------------------------------------------
// Kernel 1: proto = mean of gathered rows; collapsed GCN (rows identical so
// A@X is identity on equal-row matrices) -> single-vector 3-layer MLP -> v3.
// ---------------------------------------------------------------------------
__global__ void k_proto_mlp(const float* __restrict__ cf, const int* __restrict__ sel,
                            const float* __restrict__ g1w, const float* __restrict__ g1b,
                            const float* __restrict__ g2w, const float* __restrict__ g2b,
                            const float* __restrict__ g3w, const float* __restrict__ g3b,
                            float* __restrict__ v3out) {
    __shared__ float proto[FEAT];
    __shared__ float h1[HID];
    __shared__ float h2[HID];
    const int t = threadIdx.x;  // 256 threads

    float acc = 0.f;
#pragma unroll
    for (int s = 0; s < TEMPL; ++s)
        acc += cf[(size_t)sel[s] * FEAT + t];
    proto[t] = acc * (1.0f / TEMPL);
    __syncthreads();

    if (t < HID) {
        float a = g1b[t];
        for (int d = 0; d < FEAT; ++d) a = fmaf(proto[d], g1w[d * HID + t], a);
        h1[t] = fmaxf(a, 0.f);
    }
    __syncthreads();
    if (t < HID) {
        float a = g2b[t];
        for (int d = 0; d < HID; ++d) a = fmaf(h1[d], g2w[d * HID + t], a);
        h2[t] = fmaxf(a, 0.f);
    }
    __syncthreads();
    {
        float a = g3b[t];
        for (int h = 0; h < HID; ++h) a = fmaf(h2[h], g3w[h * FEAT + t], a);
        v3out[t] = 1.f / (1.f + expf(-a));
    }
}

// ---------------------------------------------------------------------------
// Kernel 2: trig (first output). rows 0..9 = v3; rows 10.. = v3 + 0.1*noise.
// ---------------------------------------------------------------------------
__global__ void k_build_trig(const float* __restrict__ v3, const float* __restrict__ noise,
                             float* __restrict__ trig) {
    const int r = blockIdx.x;
    const int d = threadIdx.x;
    float v = v3[d];
    if (r >= TEMPL) v = fmaf(0.1f, noise[(r - TEMPL) * FEAT + d], v);
    trig[r * FEAT + d] = v;
}

// ---------------------------------------------------------------------------
// Kernel 3: pab = trig @ [w1a | w1b], M=2048 N=128 K=256, fp32 WMMA.
// One wave = one 16x16 tile. v_wmma_f32_16x16x4_f32, 64 K-steps.
// A frag (16x4): lane<16 -> row=lane, K=k,k+1; lane>=16 -> row=lane-16, K=k+2,k+3.
// B frag (4x16): mirrored (lane<16 col=lane K=k/k+1, lane>=16 K=k+2/k+3).
// C/D (16x16): lane<16 -> M=vgpr, N=lane; lane>=16 -> M=vgpr+8, N=lane-16.
// ---------------------------------------------------------------------------
__global__ void k_gemm_pab(const float* __restrict__ trig, const float* __restrict__ ew1,
                           float* __restrict__ pab) {
    const int lane  = threadIdx.x;        // wave32, one wave per block
    const int tm    = blockIdx.x;         // 0..127
    const int tn    = blockIdx.y;         // 0..7
    const int l15   = lane & 15;
    const int khalf = (lane >> 4) * 2;    // 0 or 2

    const int rowA  = tm * 16 + l15;
    const int ncol  = tn * 16 + l15;      // 0..127 (n<64 -> pa, n>=64 -> pb)
    const int bbase = (ncol < HID) ? 0 : FEAT;  // which half of ew1 (512x64)
    const int nn    = ncol & (HID - 1);

    const float* arow = trig + (size_t)rowA * FEAT + khalf;
    v8f c = {};
    for (int k = 0; k < FEAT; k += 4) {
        v2f a = *(const v2f*)(arow + k);
        const int kb = bbase + k + khalf;
        v2f b;
        b.x = ew1[kb * HID + nn];
        b.y = ew1[(kb + 1) * HID + nn];
        c = __builtin_amdgcn_wmma_f32_16x16x4_f32(false, a, false, b, (short)0, c,
                                                  false, false);
    }

    const int n     = tn * 16 + l15;
    const int mbase = tm * 16 + ((lane >> 4) ? 8 : 0);
#pragma unroll
    for (int v = 0; v < 8; ++v)
        pab[(size_t)(mbase + v) * 128 + n] = c[v];
}

// ---------------------------------------------------------------------------
// Kernel 4: edge MLP over the upper triangle. Block = row i; pa[i]+eb1 and
// ew2 cached in LDS; pb rows streamed (L2-resident, 1MB total).
// triu row-major: base(i) = i*2047 - i*(i-1)/2, p = base + (j-i-1).
// ---------------------------------------------------------------------------
__global__ void k_edges(const float* __restrict__ pab, const float* __restrict__ eb1,
                        const float* __restrict__ ew2, const float* __restrict__ eb2,
                        float* __restrict__ out) {
    __shared__ float s_pa[HID];
    __shared__ float s_w2[HID];
    __shared__ float s_b2;
    const int i = blockIdx.x;   // 0..2046
    const int t = threadIdx.x;  // 256 threads

    if (t < HID) {
        s_pa[t] = pab[i * 128 + t] + eb1[t];
        s_w2[t] = ew2[t];
    }
    if (t == 0) s_b2 = eb2[0];
    __syncthreads();

    const int base = i * (ONODES - 1) - (i * (i - 1)) / 2;
    for (int j = i + 1 + t; j < ONODES; j += 256) {
        const float* pb = pab + (size_t)j * 128 + HID;
        float acc = 0.f;
#pragma unroll
        for (int k = 0; k < HID; k += 4) {
            v4f p4 = *(const v4f*)(pb + k);
            acc = fmaf(fmaxf(s_pa[k + 0] + p4.x, 0.f), s_w2[k + 0], acc);
            acc = fmaf(fmaxf(s_pa[k + 1] + p4.y, 0.f), s_w2[k + 1], acc);
            acc = fmaf(fmaxf(s_pa[k + 2] + p4.z, 0.f), s_w2[k + 2], acc);
            acc = fmaf(fmaxf(s_pa[k + 3] + p4.w, 0.f), s_w2[k + 3], acc);
        }
        out[base + (j - i - 1)] = 1.f / (1.f + expf(-(acc + s_b2)));
    }
}

// ---------------------------------------------------------------------------
extern "C" void kernel_launch(void* const* d_in, const int* in_sizes, int n_in,
                              void* d_out, int out_size, void* d_ws, size_t ws_size,
                              hipStream_t stream) {
    const float* cf    = (const float*)d_in[0];
    const int*   sel   = (const int*)d_in[1];
    const float* noise = (const float*)d_in[2];
    const float* g1w   = (const float*)d_in[3];
    const float* g1b   = (const float*)d_in[4];
    const float* g2w   = (const float*)d_in[5];
    const float* g2b   = (const float*)d_in[6];
    const float* g3w   = (const float*)d_in[7];
    const float* g3b   = (const float*)d_in[8];
    const float* ew1   = (const float*)d_in[9];
    const float* eb1   = (const float*)d_in[10];
    const float* ew2   = (const float*)d_in[11];
    const float* eb2   = (const float*)d_in[12];

    float* out   = (float*)d_out;
    float* trig  = out;                       // 2048*256 floats
    float* edges = out + ONODES * FEAT;       // 2048*2047/2 floats

    float* v3  = (float*)d_ws;                // 256 floats
    float* pab = v3 + 256;                    // 2048*128 floats

    k_proto_mlp<<<1, 256, 0, stream>>>(cf, sel, g1w, g1b, g2w, g2b, g3w, g3b, v3);
    k_build_trig<<<ONODES, FEAT, 0, stream>>>(v3, noise, trig);
    k_gemm_pab<<<dim3(ONODES / 16, 128 / 16), 32, 0, stream>>>(trig, ew1, pab);
    k_edges<<<ONODES - 1, 256, 0, stream>>>(pab, eb1, ew2, eb2, edges);
}